// Concat4_52226802320147
// MI455X (gfx1250) — compile-verified
//
#include <hip/hip_runtime.h>

// MI455X / gfx1250, wave32.
// Shapes fixed by the reference: B=16, C=256 per input (512 concat), H*W=4096.
#define B_   16
#define C_   256
#define CT   512          // total channels after concat
#define HW   4096         // H*W
typedef __attribute__((ext_vector_type(2))) float v2f;
typedef __attribute__((ext_vector_type(4))) float v4f;
typedef __attribute__((ext_vector_type(8))) float v8f;
typedef int v4i __attribute__((vector_size(4 * sizeof(int))));

// gfx1250 async global<->LDS copy path (ASYNCcnt-tracked), with fallback.
// Probed signature (rounds 2-3 stderr):
//   __builtin_amdgcn_global_load_async_to_lds_b128(v4i* g, __shared__ v4i* l,
//                                                  imm int offset, imm int cpol)
#if defined(__gfx1250__) &&                                                    \
    __has_builtin(__builtin_amdgcn_global_load_async_to_lds_b128) &&           \
    __has_builtin(__builtin_amdgcn_global_store_async_from_lds_b128) &&        \
    __has_builtin(__builtin_amdgcn_s_wait_asynccnt)
#define USE_ASYNC_COPY 1
typedef __attribute__((address_space(3))) v4i LV4I;
#endif

__device__ __forceinline__ const float* plane_ptr(const float* __restrict__ x1,
                                                  const float* __restrict__ x2,
                                                  int b, int c) {
    return (c < C_) ? (x1 + (size_t)(b * C_ + c) * HW)
                    : (x2 + (size_t)(b * C_ + (c - C_)) * HW);
}

// ---------------------------------------------------------------------------
// Kernel 1: per-(b,c) plane sums via V_WMMA_F32_16X16X4_F32.
// A = ones(16x4); B operand carries 64 distinct plane elements per issue.
// D[m][n] += sum_k B[k][n]; summing ALL 16 columns of D row 0 (lanes 0-15,
// acc VGPR 0 -- documented C/D layout) gives the total independent of the
// element->slot mapping. So each lane loads 4 CONSECUTIVE floats with one
// global_load_b128 (512 B/wave, fully coalesced) and feeds two WMMAs.
// One wave per plane; 8 waves per 256-thread block. This pass also pulls the
// whole 128 MiB input into the 192 MB L2 for the gather pass.
// ---------------------------------------------------------------------------
__global__ __launch_bounds__(256) void pool_kernel(const float* __restrict__ x1,
                                                   const float* __restrict__ x2,
                                                   float* __restrict__ pooled) {
    const int wave = threadIdx.x >> 5;
    const int lane = threadIdx.x & 31;
    const int row  = blockIdx.x * 8 + wave;        // [0, B_*CT)
    const int b    = row >> 9;                     // /512
    const int c    = row & (CT - 1);
    const float* __restrict__ src = plane_ptr(x1, x2, b, c);

    v2f a;  a.x = 1.0f; a.y = 1.0f;                // ones A-matrix (all lanes)
    v8f acc = {};

    #pragma unroll 2
    for (int i = 0; i < HW / 128; ++i) {           // 128 floats per wave-iter
        const v4f q = *(const v4f*)(src + i * 128 + lane * 4);
        v2f b0; b0.x = q.x; b0.y = q.y;
        v2f b1; b1.x = q.z; b1.y = q.w;
        acc = __builtin_amdgcn_wmma_f32_16x16x4_f32(
            false, a, false, b0, (short)0, acc, false, false);
        acc = __builtin_amdgcn_wmma_f32_16x16x4_f32(
            false, a, false, b1, (short)0, acc, false, false);
    }
    // acc[0]: lanes 0-15 = D[0][0..15]; reduce the 16 columns.
    float s = acc[0];
    s += __shfl_xor(s, 1, 32);
    s += __shfl_xor(s, 2, 32);
    s += __shfl_xor(s, 4, 32);
    s += __shfl_xor(s, 8, 32);
    if (lane == 0) pooled[row] = s;                // sum ~ mean * HW, same order
}

// ---------------------------------------------------------------------------
// Kernel 2: stable descending rank (matches jnp.argsort(-pooled) stability).
// perm[b][rank] = c. One block per batch, 512 threads, O(512^2) counting.
// ---------------------------------------------------------------------------
__global__ __launch_bounds__(512) void rank_kernel(const float* __restrict__ pooled,
                                                   int* __restrict__ perm) {
    __shared__ float p[CT];
    const int b = blockIdx.x;
    const int c = threadIdx.x;
    const float v = pooled[b * CT + c];
    p[c] = v;
    __syncthreads();
    int rank = 0;
    #pragma unroll 8
    for (int j = 0; j < CT; ++j) {
        const float pj = p[j];
        rank += (pj > v) || (pj == v && j < c);
    }
    perm[b * CT + rank] = c;
}

// ---------------------------------------------------------------------------
// Kernel 3: copy planes rank 0..k-2 to the output. grid = B*(k-1)*4, each
// block moves 4 KB. Preferred path: CDNA5 async global->LDS->global DMA
// (ASYNCcnt); each wave stages and stores only its own 512 B, so only an
// s_wait_asynccnt between load and store is needed (no block barrier).
// Fallback: b128 load + nontemporal b128 store (keeps inputs L2-resident).
// ---------------------------------------------------------------------------
__global__ __launch_bounds__(256) void gather_kernel(const float* __restrict__ x1,
                                                     const float* __restrict__ x2,
                                                     const int* __restrict__ perm,
                                                     float* __restrict__ out, int k) {
    const int kc  = k - 1;
    const int pb  = blockIdx.x & 3;                // 4 blocks per plane
    const int t   = blockIdx.x >> 2;
    const int r   = t % kc;
    const int b   = t / kc;
    const int c   = perm[b * CT + r];
    const v4f* __restrict__ src = (const v4f*)plane_ptr(x1, x2, b, c);
    v4f* __restrict__ dst = (v4f*)(out + (size_t)(b * k + r) * HW);
    const int pix = pb * 256 + threadIdx.x;        // float4 index in [0,1024)

#ifdef USE_ASYNC_COPY
    __shared__ __align__(16) float stage[256 * 4]; // 4 KB staging
    // as3 pointer value == LDS byte offset == low 32 bits of the flat address
    // of a __shared__ object.
    LV4I* lds = (LV4I*)(unsigned)(uintptr_t)&stage[threadIdx.x * 4];
    __builtin_amdgcn_global_load_async_to_lds_b128(
        (v4i*)(src + pix), lds, /*offset=*/0, /*cpol=*/0);
    __builtin_amdgcn_s_wait_asynccnt(0);           // own data only: no barrier
    __builtin_amdgcn_global_store_async_from_lds_b128(
        (v4i*)(dst + pix), lds, /*offset=*/0, /*cpol=*/0);
    // s_endpgm performs an implicit wait-idle, draining ASYNCcnt.
#else
    __builtin_nontemporal_store(src[pix], dst + pix);
#endif
}

// ---------------------------------------------------------------------------
// Kernel 4: out[b, k-1, :] = sum of planes rank k-1..CT-1 (257 planes).
// Deterministic (no atomics) -> safe for graph replay revalidation.
// grid = B*16 blocks, 256 threads; perm slice staged in LDS.
// ---------------------------------------------------------------------------
__global__ __launch_bounds__(256) void tail_kernel(const float* __restrict__ x1,
                                                   const float* __restrict__ x2,
                                                   const int* __restrict__ perm,
                                                   float* __restrict__ out, int k) {
    __shared__ int ch[CT];
    const int b   = blockIdx.x >> 4;
    const int pix = (blockIdx.x & 15) * 256 + threadIdx.x;   // [0, HW)
    const int n   = CT - (k - 1);                            // 257 planes
    for (int i = threadIdx.x; i < n; i += 256)
        ch[i] = perm[b * CT + (k - 1) + i];
    __syncthreads();

    float s = 0.0f;
    for (int i = 0; i < n; ++i) {
        const float* __restrict__ src = plane_ptr(x1, x2, b, ch[i]);
        s += src[pix];                                       // coalesced, L2 hits
    }
    __builtin_nontemporal_store(s, out + (size_t)(b * k + (k - 1)) * HW + pix);
}

// ---------------------------------------------------------------------------
extern "C" void kernel_launch(void* const* d_in, const int* in_sizes, int n_in,
                              void* d_out, int out_size, void* d_ws, size_t ws_size,
                              hipStream_t stream) {
    const float* x1 = (const float*)d_in[0];
    const float* x2 = (const float*)d_in[1];
    float* out = (float*)d_out;

    // k lives on device (scalar input); recover it from out_size instead of a
    // graph-capture-illegal memcpy:  out = [B, k, H, W].
    const int k = out_size / (B_ * HW);            // 256 for this reference

    float* pooled = (float*)d_ws;                            // 16*512 floats
    int*   perm   = (int*)((char*)d_ws + B_ * CT * sizeof(float)); // 16*512 ints

    pool_kernel<<<(B_ * CT) / 8, 256, 0, stream>>>(x1, x2, pooled);
    rank_kernel<<<B_, CT, 0, stream>>>(pooled, perm);
    if (k > 1)
        gather_kernel<<<B_ * (k - 1) * 4, 256, 0, stream>>>(x1, x2, perm, out, k);
    tail_kernel<<<B_ * 16, 256, 0, stream>>>(x1, x2, perm, out, k);
}